// Pretrained_LWPT_Decomposition_27960237096988
// MI455X (gfx1250) — compile-verified
//
#include <hip/hip_runtime.h>
#include <cstdint>

constexpr int   T0    = 4096;   // time samples per (b,c) row
constexpr int   NCH   = 32;
constexpr int   BLOCK = 256;    // 8 wave32
constexpr float SLOPE = 10.0f;

struct F4 { float a, b, c, d; };

__device__ __forceinline__ float sigm(float z) {
    return 1.0f / (1.0f + __expf(-z));
}

// One DWT level: read N samples from LDS `in`, write N/2 `lo` back to LDS,
// stream gated detail coefficients straight to global (non-temporal).
__device__ __forceinline__ void dwt_level(const float* __restrict__ in, int N,
                                          float* __restrict__ lo_out,
                                          float* __restrict__ det_out,
                                          F4 hl, F4 hh, float bia, int tid) {
    const int M = N >> 1;
    for (int t = tid; t < M; t += BLOCK) {
        const int i0 = 2 * t;
        int i2 = i0 + 2; if (i2 >= N) i2 -= N;   // circular pad (only last t wraps)
        int i3 = i0 + 3; if (i3 >= N) i3 -= N;
        const float a0 = in[i0];
        const float a1 = in[i0 + 1];
        const float a2 = in[i2];
        const float a3 = in[i3];
        const float lo = __fmaf_rn(hl.a, a0, __fmaf_rn(hl.b, a1,
                         __fmaf_rn(hl.c, a2, hl.d * a3)));
        const float hi = __fmaf_rn(hh.a, a0, __fmaf_rn(hh.b, a1,
                         __fmaf_rn(hh.c, a2, hh.d * a3)));
        lo_out[t] = lo;
        const float g = sigm(SLOPE * (hi - bia)) + sigm(-SLOPE * (hi + bia));
        __builtin_nontemporal_store(hi * g, det_out + t);
    }
}

__global__ __launch_bounds__(BLOCK)
void lwpt_decomp_kernel(const float* __restrict__ x,
                        const float* __restrict__ h_lo,
                        const float* __restrict__ bias,
                        float* __restrict__ out) {
    __shared__ float smem[T0 + T0 / 2];           // 24 KB: row + ping-pong
    float* bufA = smem;
    float* bufB = smem + T0;

    const int row = blockIdx.x;                   // row = b*32 + c
    const int ch  = row & (NCH - 1);
    const int tid = threadIdx.x;

    const float* xrow = x   + (size_t)row * T0;
    float*       orow = out + (size_t)row * T0;

    // ---- async DMA of the 16KB row into LDS (CDNA5 GLOBAL_LOAD_ASYNC_TO_LDS) ----
    {
        const uint64_t gbase = (uint64_t)(uintptr_t)xrow;
        // low 32 bits of a flat shared pointer are the LDS byte address
        const unsigned lbase = (unsigned)(uintptr_t)smem;
        #pragma unroll
        for (int it = 0; it < (T0 * 4) / (BLOCK * 16); ++it) {   // 4 iters of b128/lane
            const unsigned boff  = (unsigned)(tid + it * BLOCK) * 16u;
            const unsigned laddr = lbase + boff;
            asm volatile("global_load_async_to_lds_b128 %0, %1, %2"
                         :: "v"(laddr), "v"(boff), "s"(gbase)
                         : "memory");
        }
        asm volatile("s_wait_asynccnt 0" ::: "memory");
    }

    // ---- per-level filters / bias (block-uniform, 20 floats, L2-resident) ----
    F4 hl[4], hh[4];
    float bi[4];
    #pragma unroll
    for (int l = 0; l < 4; ++l) {
        const float* h = h_lo + ((size_t)ch * 4 + l) * 4;
        const float h0 = h[0], h1 = h[1], h2 = h[2], h3 = h[3];
        hl[l] = {h0, h1, h2, h3};
        hh[l] = {h3, -h2, h1, -h0};               // sign[k] * h[3-k]
        bi[l] = bias[ch * 4 + l];
    }

    __syncthreads();                              // LDS row visible to all waves

    // ---- 4 fused levels, all intermediates in LDS ----
    dwt_level(bufA, 4096, bufB, orow + 2048, hl[0], hh[0], bi[0], tid);
    __syncthreads();
    dwt_level(bufB, 2048, bufA, orow + 1024, hl[1], hh[1], bi[1], tid);
    __syncthreads();
    dwt_level(bufA, 1024, bufB, orow + 512,  hl[2], hh[2], bi[2], tid);
    __syncthreads();
    dwt_level(bufB, 512,  bufA, orow + 256,  hl[3], hh[3], bi[3], tid);
    __syncthreads();

    // final approximation coefficients: out[0:256]
    __builtin_nontemporal_store(bufA[tid], orow + tid);
}

extern "C" void kernel_launch(void* const* d_in, const int* in_sizes, int n_in,
                              void* d_out, int out_size, void* d_ws, size_t ws_size,
                              hipStream_t stream) {
    const float* x    = (const float*)d_in[0];   // (128, 32, 4096)
    const float* h_lo = (const float*)d_in[1];   // (32, 4, 4)
    const float* bias = (const float*)d_in[2];   // (32, 4)
    float* out        = (float*)d_out;           // (128, 32, 4096)

    const int rows = in_sizes[0] / T0;           // 128*32 = 4096 workgroups
    lwpt_decomp_kernel<<<rows, BLOCK, 0, stream>>>(x, h_lo, bias, out);
}